// CombinedLoss_34308198761066
// MI455X (gfx1250) — compile-verified
//
#include <hip/hip_runtime.h>
#include <hip/hip_bf16.h>
#include <math.h>

// Problem geometry (from reference): B=8, C=1, H=256, W=256, f32 everywhere.
#define BATCH 8
#define HGT   256
#define WDT   256
#define NELEM (BATCH * HGT * WDT)   // 524288
#define BIGD  512                   // 2*W cap used by reference row-EDT
#define WTILE 16                    // columns per block in the column-EDT pass

typedef float v2f __attribute__((ext_vector_type(2)));
typedef float v8f __attribute__((ext_vector_type(8)));

// ---------------------------------------------------------------------------
// Wave32 sum reduction through the matrix pipe.
// A = {x, 0} per lane -> A[m][k] holds x_m / x_{m+16} at unique slots.
// B = all ones (layout-independent). D[m][n] = x_m + x_{m+16} for every n.
// Lane's 8 C/D VGPRs cover rows {0..7} (lanes 0-15) or {8..15} (lanes 16-31),
// so sum(own 8 regs) + xor-16 partner = full 32-lane total, exact in f32.
// EXEC must be all 1s at the call site (callers guarantee full waves).
// ---------------------------------------------------------------------------
__device__ __forceinline__ float wave_sum32(float x) {
#if __has_builtin(__builtin_amdgcn_wmma_f32_16x16x4_f32)
  v2f a; a[0] = x;    a[1] = 0.0f;
  v2f b; b[0] = 1.0f; b[1] = 1.0f;
  v8f c = {};
  c = __builtin_amdgcn_wmma_f32_16x16x4_f32(
      /*neg_a=*/false, a, /*neg_b=*/false, b,
      /*c_mod=*/(short)0, c, /*reuse_a=*/false, /*reuse_b=*/false);
  float s = c[0] + c[1] + c[2] + c[3] + c[4] + c[5] + c[6] + c[7];
  s += __shfl_xor(s, 16, 32);
  return s;
#else
  for (int off = 16; off > 0; off >>= 1) x += __shfl_xor(x, off, 32);
  return x;
#endif
}

// ---------------------------------------------------------------------------
// Kernel 1: per-row 1-D EDT for mask t (zeros = t<0.5) and mask 1-t
// (zeros = t>=0.5). Writes squared distances (f32) capped at BIGD.
// One thread owns one full row (serial two-pass scan, data is L2-resident).
// ---------------------------------------------------------------------------
__global__ void row_edt_kernel(const float* __restrict__ tgt,
                               float* __restrict__ dP,
                               float* __restrict__ dN) {
  int r = blockIdx.x * blockDim.x + threadIdx.x;
  if (r >= BATCH * HGT) return;
  const float* trow = tgt + (size_t)r * WDT;
  float* pr = dP + (size_t)r * WDT;
  float* nr = dN + (size_t)r * WDT;

  int lastP = -BIGD, lastN = -BIGD;
  for (int x = 0; x < WDT; ++x) {
    bool zP = (trow[x] < 0.5f);
    if (zP) lastP = x; else lastN = x;
    pr[x] = (float)(x - lastP);   // left distance (>= BIGD if none yet)
    nr[x] = (float)(x - lastN);
  }
  int nextP = 1 << 20, nextN = 1 << 20;
  for (int x = WDT - 1; x >= 0; --x) {
    bool zP = (trow[x] < 0.5f);
    if (zP) nextP = x; else nextN = x;
    float d = fminf(fminf(pr[x], (float)(nextP - x)), (float)BIGD);
    pr[x] = d * d;
    d = fminf(fminf(nr[x], (float)(nextN - x)), (float)BIGD);
    nr[x] = d * d;
  }
}

// ---------------------------------------------------------------------------
// Kernel 2: column pass of the 2-D EDT + boundary-weight map.
// Block = (batch b, 16-column tile). LDS stages d1^2 columns for both masks;
// thread i computes min_j (d1sq[j][w] + (i-j)^2) for 16 columns, then
// bw = sigmoid(-(dist - THETA0)/THETA).
// ---------------------------------------------------------------------------
__global__ void col_edt_bw_kernel(const float* __restrict__ tgt,
                                  const float* __restrict__ dP,
                                  const float* __restrict__ dN,
                                  float* __restrict__ bw) {
  __shared__ float sP[HGT * WTILE];
  __shared__ float sN[HGT * WTILE];

  const int b  = blockIdx.y;
  const int wb = blockIdx.x * WTILE;
  const size_t base = (size_t)b * HGT * WDT;

  // Stage: thread j loads 16 consecutive floats of row j for both maps.
  {
    const int j = threadIdx.x;
    const size_t g = base + (size_t)j * WDT + wb;
#pragma unroll
    for (int w = 0; w < WTILE; ++w) {
      sP[j * WTILE + w] = dP[g + w];
      sN[j * WTILE + w] = dN[g + w];
    }
  }
  __syncthreads();

  const int i = threadIdx.x;
  float mP[WTILE], mN[WTILE];
#pragma unroll
  for (int w = 0; w < WTILE; ++w) { mP[w] = 1e30f; mN[w] = 1e30f; }

  for (int j = 0; j < HGT; ++j) {
    const float dij = (float)(i - j);
    const float off2 = dij * dij;
#pragma unroll
    for (int w = 0; w < WTILE; ++w) {
      mP[w] = fminf(mP[w], sP[j * WTILE + w] + off2);
      mN[w] = fminf(mN[w], sN[j * WTILE + w] + off2);
    }
  }

  const size_t o = base + (size_t)i * WDT + wb;
#pragma unroll
  for (int w = 0; w < WTILE; ++w) {
    const float tv = tgt[o + w];
    const float dist = (tv > 0.5f) ? sqrtf(mP[w]) : sqrtf(mN[w]);
    // sigmoid(-(|dist| - 3)/5) = 1 / (1 + exp((dist - 3)/5))
    bw[o + w] = 1.0f / (1.0f + expf((dist - 3.0f) * 0.2f));
  }
}

// ---------------------------------------------------------------------------
// Kernel 3: elementwise BCE/sigmoid terms + 9 global sums.
// acc[0]=bce_main  acc[1]=bce*bw main  acc[2]=p_main  acc[3]=p*t main
// acc[4]=t         acc[5]=bce_aux   acc[6]=bce*bw aux acc[7]=p_aux acc[8]=p*t aux
// 65536 threads x exactly 8 elements -> EXEC all-1s at the WMMA reduction.
// ---------------------------------------------------------------------------
__global__ void loss_partial_kernel(const float* __restrict__ pred,
                                    const float* __restrict__ tgt,
                                    const float* __restrict__ aux,
                                    const float* __restrict__ bw,
                                    float* __restrict__ acc) {
  const int tid = blockIdx.x * blockDim.x + threadIdx.x;
  const int stride = gridDim.x * blockDim.x;

  float s0 = 0.f, s1 = 0.f, s2 = 0.f, s3 = 0.f, s4 = 0.f;
  float s5 = 0.f, s6 = 0.f, s7 = 0.f, s8 = 0.f;

  for (int i = tid; i < NELEM; i += stride) {
    const float t  = tgt[i];
    const float w  = bw[i];
    const float lm = pred[i];
    const float la = aux[i];
    const float bm = fmaxf(lm, 0.f) - lm * t + log1pf(expf(-fabsf(lm)));
    const float ba = fmaxf(la, 0.f) - la * t + log1pf(expf(-fabsf(la)));
    const float pm = 1.f / (1.f + expf(-lm));
    const float pa = 1.f / (1.f + expf(-la));
    s0 += bm; s1 += bm * w; s2 += pm; s3 += pm * t; s4 += t;
    s5 += ba; s6 += ba * w; s7 += pa; s8 += pa * t;
  }

  float part[9] = {s0, s1, s2, s3, s4, s5, s6, s7, s8};
#pragma unroll
  for (int k = 0; k < 9; ++k) {
    const float tot = wave_sum32(part[k]);   // v_wmma_f32_16x16x4_f32 path
    if ((threadIdx.x & 31) == 0) atomicAdd(&acc[k], tot);
  }
}

__global__ void init_acc_kernel(float* __restrict__ acc) {
  if (threadIdx.x < 9) acc[threadIdx.x] = 0.0f;
}

__global__ void finalize_kernel(const float* __restrict__ acc,
                                float* __restrict__ out) {
  const float n = (float)NELEM;
  const float bceM  = acc[0] / n;
  const float bndM  = acc[1] / n;
  const float diceM = 1.f - (2.f * acc[3] + 1.f) / (acc[2] + acc[4] + 1.f);
  const float bceA  = acc[5] / n;
  const float bndA  = acc[6] / n;
  const float diceA = 1.f - (2.f * acc[8] + 1.f) / (acc[7] + acc[4] + 1.f);
  out[0] = (bceM + diceM + bndM) + 0.4f * (bceA + diceA + bndA);
}

// ---------------------------------------------------------------------------
extern "C" void kernel_launch(void* const* d_in, const int* in_sizes, int n_in,
                              void* d_out, int out_size, void* d_ws, size_t ws_size,
                              hipStream_t stream) {
  (void)in_sizes; (void)n_in; (void)out_size; (void)ws_size;

  const float* pred = (const float*)d_in[0];   // predictions
  const float* tgt  = (const float*)d_in[1];   // targets (0/1 floats)
  const float* aux  = (const float*)d_in[2];   // aux_predictions
  float* out = (float*)d_out;

  // Workspace layout (floats): d1sq_pos | d1sq_neg | bw | acc[9]
  float* dP  = (float*)d_ws;
  float* dN  = dP + NELEM;
  float* bw  = dN + NELEM;
  float* acc = bw + NELEM;

  init_acc_kernel<<<1, 32, 0, stream>>>(acc);

  row_edt_kernel<<<(BATCH * HGT) / 32, 32, 0, stream>>>(tgt, dP, dN);

  dim3 grid2(WDT / WTILE, BATCH);
  col_edt_bw_kernel<<<grid2, HGT, 0, stream>>>(tgt, dP, dN, bw);

  loss_partial_kernel<<<256, 256, 0, stream>>>(pred, tgt, aux, bw, acc);

  finalize_kernel<<<1, 1, 0, stream>>>(acc, out);
}